// GatedGCNNet1_83073257439661
// MI455X (gfx1250) — compile-verified
//
#include <hip/hip_runtime.h>
#include <math.h>

typedef __attribute__((ext_vector_type(2))) float v2f;
typedef __attribute__((ext_vector_type(8))) float v8f;

#define KS64 68    // 64 + 4 pad (floats) -> kills LDS bank conflicts for b64 loads
#define KS128 132  // 128 + 4 pad

// ---------------------------------------------------------------------------
// Generic tiled GEMM: Y[ntiles*16, 64] = X[ntiles*16, K] @ W[K, 64] + bias
// 128 threads = 4 waves; wave w computes the 16x16 output tile at cols [16w,16w+16)
// using V_WMMA_F32_16X16X4_F32 (f32 in, f32 acc -> bit-compatible with reference).
// ---------------------------------------------------------------------------
template <int K, int KS>
__global__ void gemm_kernel(const float* __restrict__ X, const float* __restrict__ W,
                            const float* __restrict__ bias, float* __restrict__ Y,
                            int ntiles) {
  __shared__ alignas(16) float sW[64 * KS];  // transposed: sW[col*KS + k]
  __shared__ alignas(16) float sA[16 * KS];  // row-major tile: sA[row*KS + k]
  __shared__ float sb[64];
  const int tid  = threadIdx.x;   // 128
  const int lane = tid & 31;
  const int wv   = tid >> 5;      // 0..3

  for (int i = tid; i < 64 * K; i += 128) {   // W row-major [K][64] -> transpose
    int k = i / 64, j = i % 64;
    sW[j * KS + k] = W[i];
  }
  if (tid < 64) sb[tid] = bias[tid];
  __syncthreads();

  const int c0    = wv * 16;
  const int col   = lane & 15;
  const int rbase = (lane >> 4) << 3;         // acc rows: v + rbase
  const int kh    = (lane >> 4) << 1;         // K-half select: 0 or 2

  for (int tile = blockIdx.x; tile < ntiles; tile += gridDim.x) {
    const float4* src = (const float4*)(X + (size_t)tile * 16 * K);
    for (int i = tid; i < 16 * K / 4; i += 128) {
      int r = i / (K / 4), q = i % (K / 4);
      float4 v = src[i];
      float* dp = &sA[r * KS + q * 4];
      dp[0] = v.x; dp[1] = v.y; dp[2] = v.z; dp[3] = v.w;
    }
    __syncthreads();

    float bv = sb[c0 + col];
    v8f acc = {bv, bv, bv, bv, bv, bv, bv, bv};
    const float* Wp = sW + c0 * KS;
#pragma unroll
    for (int kb = 0; kb < K; kb += 4) {
      v2f a = *(const v2f*)(sA + col * KS + kb + kh);  // A row = lane&15
      v2f b = *(const v2f*)(Wp + col * KS + kb + kh);  // B col = lane&15
      acc = __builtin_amdgcn_wmma_f32_16x16x4_f32(false, a, false, b, (short)0, acc,
                                                  false, false);
    }
    float* out = Y + (size_t)tile * 16 * 64;
#pragma unroll
    for (int v = 0; v < 8; ++v) out[(rbase + v) * 64 + c0 + col] = acc[v];
    __syncthreads();
  }
}

// ---------------------------------------------------------------------------
// Fused edge pass (per 16-edge tile):
//   enew = e@Wc + bc  (WMMA)  + Dh[src] + Eh[dst]
//   sigma = sigmoid(enew); atomic num[dst]+=sigma*Bh[src]; den[dst]+=sigma
//   e_tmp = enew * edge_norm (stored); accumulate BN sum/sumsq per column
// 128 threads: 4 waves for WMMA, then 8 threads/edge x 8 contiguous cols each.
// ---------------------------------------------------------------------------
__global__ void edge_kernel(const float* __restrict__ Ein, const float* __restrict__ Wc,
                            const float* __restrict__ bc, const float* __restrict__ Dh,
                            const float* __restrict__ Eh, const float* __restrict__ Bh,
                            const int* __restrict__ esrc, const int* __restrict__ edst,
                            const float* __restrict__ enorm, float* __restrict__ numb,
                            float* __restrict__ denb, float* __restrict__ etmp,
                            float* __restrict__ stats /*[128]: sum,sumsq*/, int ntiles) {
  __shared__ alignas(16) float sW[64 * KS64];
  __shared__ alignas(16) float sA[16 * KS64];
  __shared__ alignas(16) float sE[16 * KS64];
  __shared__ float sb[64];
  __shared__ float sstat[128];
  __shared__ int   sidx[32];
  __shared__ float snorm[16];

  const int tid  = threadIdx.x;  // 128
  const int lane = tid & 31;
  const int wv   = tid >> 5;

  for (int i = tid; i < 64 * 64; i += 128) {
    int k = i >> 6, j = i & 63;
    sW[j * KS64 + k] = Wc[i];
  }
  if (tid < 64) sb[tid] = bc[tid];
  sstat[tid] = 0.f;
  __syncthreads();

  const int c0    = wv * 16;
  const int col   = lane & 15;
  const int rbase = (lane >> 4) << 3;
  const int kh    = (lane >> 4) << 1;
  const int ei    = tid >> 3;   // edge within tile
  const int q     = tid & 7;    // covers cols [8q, 8q+8)

  float accS[8], accQ[8];
#pragma unroll
  for (int j = 0; j < 8; ++j) { accS[j] = 0.f; accQ[j] = 0.f; }

  for (int tile = blockIdx.x; tile < ntiles; tile += gridDim.x) {
    const float4* src = (const float4*)(Ein + (size_t)tile * 16 * 64);
    for (int i = tid; i < 256; i += 128) {
      int r = i >> 4, p = i & 15;
      float4 v = src[i];
      float* dp = &sA[r * KS64 + p * 4];
      dp[0] = v.x; dp[1] = v.y; dp[2] = v.z; dp[3] = v.w;
    }
    if (tid < 16) {
      sidx[tid]      = esrc[tile * 16 + tid];
      sidx[16 + tid] = edst[tile * 16 + tid];
      snorm[tid]     = enorm[tile * 16 + tid];
    }
    __syncthreads();

    // WMMA: enew_tile = e_tile @ Wc + bc
    float bv = sb[c0 + col];
    v8f acc = {bv, bv, bv, bv, bv, bv, bv, bv};
    const float* Wp = sW + c0 * KS64;
#pragma unroll
    for (int kb = 0; kb < 64; kb += 4) {
      v2f a = *(const v2f*)(sA + col * KS64 + kb + kh);
      v2f b = *(const v2f*)(Wp + col * KS64 + kb + kh);
      acc = __builtin_amdgcn_wmma_f32_16x16x4_f32(false, a, false, b, (short)0, acc,
                                                  false, false);
    }
#pragma unroll
    for (int v = 0; v < 8; ++v) sE[(rbase + v) * KS64 + c0 + col] = acc[v];
    __syncthreads();

    // per-edge gather / sigmoid / scatter
    const int s  = sidx[ei];
    const int d  = sidx[16 + ei];
    const float en = snorm[ei];
    const float* Dr = Dh + (size_t)s * 64 + q * 8;
    const float* Er = Eh + (size_t)d * 64 + q * 8;
    const float* Br = Bh + (size_t)s * 64 + q * 8;
    float* nm = numb + (size_t)d * 64 + q * 8;
    float* dn = denb + (size_t)d * 64 + q * 8;
    float* et = etmp + ((size_t)tile * 16 + ei) * 64 + q * 8;
#pragma unroll
    for (int j = 0; j < 8; ++j) {
      float v  = sE[ei * KS64 + q * 8 + j] + Dr[j] + Er[j];
      float sg = 1.f / (1.f + expf(-v));
      atomicAdd(nm + j, sg * Br[j]);
      atomicAdd(dn + j, sg);
      float ev = v * en;
      et[j] = ev;
      accS[j] += ev;
      accQ[j] += ev * ev;
    }
    __syncthreads();
  }
#pragma unroll
  for (int j = 0; j < 8; ++j) {
    atomicAdd(&sstat[q * 8 + j], accS[j]);        // ds_add_f32
    atomicAdd(&sstat[64 + q * 8 + j], accQ[j]);
  }
  __syncthreads();
  atomicAdd(&stats[tid], sstat[tid]);
}

// h_tmp = (Ah + num/(den+1e-6)) * node_norm ; accumulate BN stats
__global__ void combine_kernel(const float* __restrict__ Ah, const float* __restrict__ numb,
                               const float* __restrict__ denb, const float* __restrict__ nnorm,
                               float* __restrict__ htmp, float* __restrict__ stats, int N) {
  __shared__ float sstat[128];
  const int tid = threadIdx.x;  // 256
  if (tid < 128) sstat[tid] = 0.f;
  __syncthreads();
  const int c = tid & 63;
  float s = 0.f, sq = 0.f;
  const size_t total = (size_t)N * 64;
  const size_t stride = (size_t)gridDim.x * 256;
  for (size_t i = (size_t)blockIdx.x * 256 + tid; i < total; i += stride) {
    float v = (Ah[i] + numb[i] / (denb[i] + 1e-6f)) * nnorm[i >> 6];
    htmp[i] = v;
    s += v; sq += v * v;
  }
  atomicAdd(&sstat[c], s);
  atomicAdd(&sstat[64 + c], sq);
  __syncthreads();
  if (tid < 128) atomicAdd(&stats[tid], sstat[tid]);
}

// BN stats -> affine coeffs:  y = a*x + b  with a=gamma*rsqrt(var+eps), b=beta-mu*a
__global__ void finalize_kernel(const float* __restrict__ stH, const float* __restrict__ stE,
                                const float* __restrict__ gh, const float* __restrict__ bh,
                                const float* __restrict__ ge, const float* __restrict__ be,
                                float* __restrict__ coef, float invN, float invE) {
  const int tid = threadIdx.x;  // 128
  if (tid < 64) {
    float mu  = stH[tid] * invN;
    float var = stH[64 + tid] * invN - mu * mu;
    float a   = gh[tid] * rsqrtf(var + 1e-5f);
    coef[tid]      = a;
    coef[64 + tid] = bh[tid] - mu * a;
  } else {
    int c = tid - 64;
    float mu  = stE[c] * invE;
    float var = stE[64 + c] * invE - mu * mu;
    float a   = ge[c] * rsqrtf(var + 1e-5f);
    coef[128 + c] = a;
    coef[192 + c] = be[c] - mu * a;
  }
}

// io += relu(a*tmp + b)   (residual + BN + ReLU, in place)
__global__ void apply_kernel(float* __restrict__ io, const float* __restrict__ tmp,
                             const float* __restrict__ coef, size_t total) {
  const int c = threadIdx.x & 63;
  const float a = coef[c], b = coef[64 + c];
  const size_t stride = (size_t)gridDim.x * 256;
  for (size_t i = (size_t)blockIdx.x * 256 + threadIdx.x; i < total; i += stride) {
    float v = a * tmp[i] + b;
    io[i] += fmaxf(v, 0.f);
  }
}

__global__ void embed_e_kernel(const float* __restrict__ ef, const float* __restrict__ We,
                               const float* __restrict__ be, float* __restrict__ e, size_t E) {
  __shared__ float w[64], b[64];
  if (threadIdx.x < 64) { w[threadIdx.x] = We[threadIdx.x]; b[threadIdx.x] = be[threadIdx.x]; }
  __syncthreads();
  const int c = threadIdx.x & 63;
  const size_t total = E * 64, stride = (size_t)gridDim.x * 256;
  for (size_t i = (size_t)blockIdx.x * 256 + threadIdx.x; i < total; i += stride)
    e[i] = ef[i >> 6] * w[c] + b[c];
}

__global__ void zero_kernel(float* __restrict__ p, size_t n) {
  const size_t stride = (size_t)gridDim.x * blockDim.x;
  for (size_t i = (size_t)blockIdx.x * blockDim.x + threadIdx.x; i < n; i += stride)
    p[i] = 0.f;
}

__global__ void readout_acc_kernel(const float* __restrict__ h, const int* __restrict__ gid,
                                   float* __restrict__ out, float* __restrict__ counts, int N) {
  const size_t total = (size_t)N * 64, stride = (size_t)gridDim.x * 256;
  for (size_t i = (size_t)blockIdx.x * 256 + threadIdx.x; i < total; i += stride) {
    size_t n = i >> 6;
    int c = (int)(i & 63);
    int g = gid[n];
    atomicAdd(&out[(size_t)g * 64 + c], h[i]);
    if (c == 0) atomicAdd(&counts[g], 1.0f);
  }
}

__global__ void readout_div_kernel(float* __restrict__ out, const float* __restrict__ counts,
                                   int G) {
  const size_t total = (size_t)G * 64, stride = (size_t)gridDim.x * 256;
  for (size_t i = (size_t)blockIdx.x * 256 + threadIdx.x; i < total; i += stride)
    out[i] /= fmaxf(counts[i >> 6], 1.0f);
}

// ---------------------------------------------------------------------------
extern "C" void kernel_launch(void* const* d_in, const int* in_sizes, int n_in,
                              void* d_out, int out_size, void* d_ws, size_t ws_size,
                              hipStream_t stream) {
  const float* nodes_feat = (const float*)d_in[0];
  const float* edges_feat = (const float*)d_in[1];
  const float* nnorm      = (const float*)d_in[2];
  const float* enorm      = (const float*)d_in[3];
  const float* W_h = (const float*)d_in[4];
  const float* b_h = (const float*)d_in[5];
  const float* W_e = (const float*)d_in[6];
  const float* b_e = (const float*)d_in[7];
  const float* Wa = (const float*)d_in[8];
  const float* ba = (const float*)d_in[9];
  const float* Wb = (const float*)d_in[10];
  const float* bb = (const float*)d_in[11];
  const float* Wc = (const float*)d_in[12];
  const float* bc = (const float*)d_in[13];
  const float* Wd = (const float*)d_in[14];
  const float* bd = (const float*)d_in[15];
  const float* Wee = (const float*)d_in[16];
  const float* bee = (const float*)d_in[17];
  const float* gamma_h = (const float*)d_in[18];
  const float* beta_h  = (const float*)d_in[19];
  const float* gamma_e = (const float*)d_in[20];
  const float* beta_e  = (const float*)d_in[21];
  const int* esrc = (const int*)d_in[22];
  const int* edst = (const int*)d_in[23];
  const int* gid  = (const int*)d_in[24];

  const int N = in_sizes[0] / 128;
  const int E = in_sizes[1];
  const int L = in_sizes[8] / (64 * 64);
  const int G = out_size / 64;
  const size_t ND = (size_t)N * 64;
  const size_t ED = (size_t)E * 64;

  char* wsp = (char*)d_ws;
  size_t off = 0;
  auto alloc = [&](size_t bytes) -> float* {
    float* p = (float*)(wsp + off);
    off += (bytes + 255) & ~(size_t)255;
    return p;
  };
  float* h     = alloc(ND * 4);
  float* e     = alloc(ED * 4);
  float* etmp  = alloc(ED * 4);
  float* Ahb   = alloc(ND * 4);
  float* Bhb   = alloc(ND * 4);
  float* Dhb   = alloc(ND * 4);
  float* Ehb   = alloc(ND * 4);
  float* numb  = alloc(ND * 4);
  float* denb  = alloc(ND * 4);
  float* htmp  = alloc(ND * 4);
  float* stats = alloc(256 * 4);  // [0..127]=h sum/sumsq, [128..255]=e sum/sumsq
  float* coef  = alloc(256 * 4);  // [0..127]=h a/b, [128..255]=e a/b
  float* cnts  = alloc((size_t)G * 4);

  const int ntN = N / 16;  // 3125
  const int ntE = E / 16;  // 50000
  float* outf = (float*)d_out;

  // Input embeddings
  gemm_kernel<128, KS128><<<512, 128, 0, stream>>>(nodes_feat, W_h, b_h, h, ntN);
  embed_e_kernel<<<1024, 256, 0, stream>>>(edges_feat, W_e, b_e, e, (size_t)E);

  for (int l = 0; l < L; ++l) {
    const size_t wo = (size_t)l * 64 * 64, bo = (size_t)l * 64;
    zero_kernel<<<512, 256, 0, stream>>>(numb, ND);
    zero_kernel<<<512, 256, 0, stream>>>(denb, ND);
    zero_kernel<<<1, 256, 0, stream>>>(stats, 256);

    gemm_kernel<64, KS64><<<512, 128, 0, stream>>>(h, Wa + wo, ba + bo, Ahb, ntN);
    gemm_kernel<64, KS64><<<512, 128, 0, stream>>>(h, Wb + wo, bb + bo, Bhb, ntN);
    gemm_kernel<64, KS64><<<512, 128, 0, stream>>>(h, Wd + wo, bd + bo, Dhb, ntN);
    gemm_kernel<64, KS64><<<512, 128, 0, stream>>>(h, Wee + wo, bee + bo, Ehb, ntN);

    edge_kernel<<<2048, 128, 0, stream>>>(e, Wc + wo, bc + bo, Dhb, Ehb, Bhb, esrc, edst,
                                          enorm, numb, denb, etmp, stats + 128, ntE);

    combine_kernel<<<1024, 256, 0, stream>>>(Ahb, numb, denb, nnorm, htmp, stats, N);
    finalize_kernel<<<1, 128, 0, stream>>>(stats, stats + 128, gamma_h + bo, beta_h + bo,
                                           gamma_e + bo, beta_e + bo, coef,
                                           1.f / (float)N, 1.f / (float)E);
    apply_kernel<<<1024, 256, 0, stream>>>(h, htmp, coef, ND);
    apply_kernel<<<2048, 256, 0, stream>>>(e, etmp, coef + 128, ED);
  }

  zero_kernel<<<64, 256, 0, stream>>>(outf, (size_t)out_size);
  zero_kernel<<<2, 256, 0, stream>>>(cnts, (size_t)G);
  readout_acc_kernel<<<1024, 256, 0, stream>>>(h, gid, outf, cnts, N);
  readout_div_kernel<<<64, 256, 0, stream>>>(outf, cnts, G);
}